// DOSAConLoss_23287312679698
// MI455X (gfx1250) — compile-verified
//
#include <hip/hip_runtime.h>
#include <math.h>

typedef __attribute__((ext_vector_type(2))) float v2f;
typedef __attribute__((ext_vector_type(8))) float v8f;

#define GAMMA_LOCAL 2.5f
#define ALPHA_W     1.2f
#define DELTA_M     1.0f
#define TAU_T       0.3f
#define LAMBDA_C    0.5f
#define EPSF        1e-7f
#define C4PI2       0.4052847345693511f  /* 4/pi^2 */

__device__ __forceinline__ float plain_iou(float x1, float y1, float w1, float h1,
                                           float x2, float y2, float w2, float h2) {
    float b1x1 = x1 - 0.5f * w1, b1x2 = x1 + 0.5f * w1;
    float b1y1 = y1 - 0.5f * h1, b1y2 = y1 + 0.5f * h1;
    float b2x1 = x2 - 0.5f * w2, b2x2 = x2 + 0.5f * w2;
    float b2y1 = y2 - 0.5f * h2, b2y2 = y2 + 0.5f * h2;
    float iw = fmaxf(fminf(b1x2, b2x2) - fmaxf(b1x1, b2x1), 0.f);
    float ih = fmaxf(fminf(b1y2, b2y2) - fmaxf(b1y1, b2y1), 0.f);
    float inter = iw * ih;
    float uni = w1 * h1 + w2 * h2 - inter + EPSF;
    return inter / uni;
}

// ---------------------------------------------------------------------------
// Kernel 1: per-row CIoU factors. Because saf_ciou broadcasts (N,1)/(N,) into
// an (N,N) outer product, the mean factorizes exactly:
//   loss_loc = mean_i(num_i) * mean_j(inv_j)
// Each block writes partial sums of both to ws.
// ---------------------------------------------------------------------------
__global__ void loc_partials_kernel(const float* __restrict__ pb,
                                    const float* __restrict__ tb,
                                    const float* __restrict__ dens,
                                    float* __restrict__ ws, int N) {
    __shared__ float sNum[256];
    __shared__ float sInv[256];
    const int t = threadIdx.x;
    const int i = blockIdx.x * 256 + t;
    float num = 0.f, inv = 0.f;
    if (i < N) {
        float x1 = pb[4 * i + 0], y1 = pb[4 * i + 1], w1 = pb[4 * i + 2], h1 = pb[4 * i + 3];
        float x2 = tb[4 * i + 0], y2 = tb[4 * i + 1], w2 = tb[4 * i + 2], h2 = tb[4 * i + 3];
        float b1x1 = x1 - 0.5f * w1, b1x2 = x1 + 0.5f * w1;
        float b1y1 = y1 - 0.5f * h1, b1y2 = y1 + 0.5f * h1;
        float b2x1 = x2 - 0.5f * w2, b2x2 = x2 + 0.5f * w2;
        float b2y1 = y2 - 0.5f * h2, b2y2 = y2 + 0.5f * h2;
        float iw = fmaxf(fminf(b1x2, b2x2) - fmaxf(b1x1, b2x1), 0.f);
        float ih = fmaxf(fminf(b1y2, b2y2) - fmaxf(b1y1, b2y1), 0.f);
        float inter = iw * ih;
        float uni = w1 * h1 + w2 * h2 - inter + EPSF;
        float iou = inter / uni;
        float cw = fmaxf(b1x2, b2x2) - fminf(b1x1, b2x1);
        float ch = fmaxf(b1y2, b2y2) - fminf(b1y1, b2y1);
        float c2 = cw * cw + ch * ch + EPSF;
        float dx = b2x1 + b2x2 - b1x1 - b1x2;
        float dy = b2y1 + b2y2 - b1y1 - b1y2;
        float rho2 = (dx * dx + dy * dy) * 0.25f;
        float dat = atanf(w2 / h2) - atanf(w1 / h1);
        float v = C4PI2 * dat * dat;
        float alpha = v / (v - iou + (1.f + EPSF));
        float ciou = iou - (rho2 / c2 + v * alpha);
        float hard = 1.f / (1.f + expf(-5.f * (0.5f - ciou)));
        num = (1.f + ALPHA_W * dens[i]) * hard * powf(1.f - ciou, GAMMA_LOCAL);
        inv = 1.f / (w2 * h2 + 1e-7f);
    }
    sNum[t] = num;
    sInv[t] = inv;
    __syncthreads();
    for (int s = 128; s > 0; s >>= 1) {
        if (t < s) { sNum[t] += sNum[t + s]; sInv[t] += sInv[t + s]; }
        __syncthreads();
    }
    if (t == 0) {
        ws[2 * blockIdx.x + 0] = sNum[0];
        ws[2 * blockIdx.x + 1] = sInv[0];
    }
}

// ---------------------------------------------------------------------------
// Kernel 2 (one 256-thread block = 8 waves): contrastive term via fp32 WMMA
// Gram diagonal + final combine. Each wave owns 16 pairs; the diff rows
// d[m][k] = e_i[m][k] - e_j[m][k] form a 16xK matrix and we accumulate
// D*D^T with V_WMMA_F32_16X16X4_F32. For this shape the A layout
// (lane=M, K=2*half+vgpr) and B layout (lane=N, K=2*half+vgpr) are mutual
// transposes, so identical registers for A and B yield the Gram matrix.
// The K-loop runs over Dv = D & ~3 with NO per-element guards (no EXEC
// manipulation, clean global_load_b64s); any K tail is added scalar-wise to
// the extracted diagonal. Invalid pairs (p >= P) read pair 0's rows — their
// garbage only lands on off-diagonals / masked lanes, never in the result.
// ---------------------------------------------------------------------------
__global__ void contrast_finalize_kernel(const float* __restrict__ pb,
                                         const float* __restrict__ emb,
                                         const int* __restrict__ idx,
                                         const float* __restrict__ ws,
                                         float* __restrict__ out,
                                         int N, int D, int P, int nblk) {
    const int t = threadIdx.x;
    const int lane = t & 31;
    const int wave = t >> 5;      // 0..7
    const int half = lane >> 4;   // K-half selector
    const int m = lane & 15;      // pair column within the wave's 16-pair tile
    const int Dv = D & ~3;        // WMMA-covered K extent
    float acc = 0.f;

    const int Ppad = (P + 127) & ~127;
    for (int base = wave * 16; base < Ppad; base += 128) {
        int p = base + m;
        float pmask = (p < P) ? 1.f : 0.f;
        int pc = (p < P) ? p : 0;
        int i0 = idx[2 * pc + 0];
        int i1 = idx[2 * pc + 1];
        const float* ei = emb + (size_t)i0 * (size_t)D;
        const float* ej = emb + (size_t)i1 * (size_t)D;

        // This lane streams K = k0 + 2*half + {0,1}; 8B-aligned since D%4==0
        // on this path (row starts at i*D floats, D multiple of 4).
        const float2* ei2 = (const float2*)(ei + 2 * half);
        const float2* ej2 = (const float2*)(ej + 2 * half);

        v8f c = {0.f, 0.f, 0.f, 0.f, 0.f, 0.f, 0.f, 0.f};
#pragma unroll 4
        for (int k0 = 0; k0 < Dv; k0 += 4) {
            float2 a0 = *ei2;
            float2 b0 = *ej2;
            ei2 += 2;             // advance 4 floats
            ej2 += 2;
            v2f a;
            a.x = a0.x - b0.x;
            a.y = a0.y - b0.y;
            // Gram accumulate: D_tile = A(16x4) * A^T(4x16) + C
            c = __builtin_amdgcn_wmma_f32_16x16x4_f32(
                    false, a, false, a, (short)0, c, false, false);
        }

        // Diagonal element (M==N==m) lives in VGPR r = m&7 on lanes where
        // (m>>3)==half (C/D layout: row = r + 8*half, col = lane%16).
        float dist2 = c[0];
#pragma unroll
        for (int r = 1; r < 8; ++r) dist2 = ((m & 7) == r) ? c[r] : dist2;
        float own = ((m >> 3) == half) ? 1.f : 0.f;

        // Scalar K tail (empty when D%4==0); uniform trip count, no EXEC games.
        for (int k = Dv; k < D; ++k) {
            float d = ei[k] - ej[k];
            dist2 += d * d;
        }

        // Plain-IoU gate for this pair (pred boxes i0 vs i1).
        float iou = plain_iou(pb[4 * i0 + 0], pb[4 * i0 + 1], pb[4 * i0 + 2], pb[4 * i0 + 3],
                              pb[4 * i1 + 0], pb[4 * i1 + 1], pb[4 * i1 + 2], pb[4 * i1 + 3]);
        float gate = (iou > TAU_T) ? 1.f : 0.f;
        float dd = fmaxf(0.f, DELTA_M - sqrtf(dist2));
        acc += own * pmask * gate * dd * dd;
    }

    __shared__ float sred[256];
    sred[t] = acc;
    __syncthreads();
    for (int s = 128; s > 0; s >>= 1) {
        if (t < s) sred[t] += sred[t + s];
        __syncthreads();
    }
    if (t == 0) {
        float sumNum = 0.f, sumInv = 0.f;
        for (int b = 0; b < nblk; ++b) {
            sumNum += ws[2 * b + 0];
            sumInv += ws[2 * b + 1];
        }
        float Nf = (float)N;
        float loss_loc = (sumNum / Nf) * (sumInv / Nf);
        float loss_con = LAMBDA_C * sred[0] / ((float)P + 1e-7f);
        out[0] = loss_loc + loss_con;
    }
}

extern "C" void kernel_launch(void* const* d_in, const int* in_sizes, int n_in,
                              void* d_out, int out_size, void* d_ws, size_t ws_size,
                              hipStream_t stream) {
    const float* pb   = (const float*)d_in[0];  // pred_boxes   (N,4) f32
    const float* tb   = (const float*)d_in[1];  // target_boxes (N,4) f32
    const float* emb  = (const float*)d_in[2];  // embeddings   (N,D) f32
    const float* dens = (const float*)d_in[3];  // density_map  (N,)  f32
    const int*   idx  = (const int*)d_in[4];    // indices      (P,2) i32

    const int N = in_sizes[0] / 4;
    const int D = (N > 0) ? (in_sizes[2] / N) : 0;
    const int P = in_sizes[4] / 2;

    float* ws  = (float*)d_ws;
    float* out = (float*)d_out;

    const int nblk = (N + 255) / 256;
    loc_partials_kernel<<<nblk, 256, 0, stream>>>(pb, tb, dens, ws, N);
    contrast_finalize_kernel<<<1, 256, 0, stream>>>(pb, emb, idx, ws, out, N, D, P, nblk);
}